// SelfAttention_27522150432840
// MI455X (gfx1250) — compile-verified
//
#include <hip/hip_runtime.h>
#include <math.h>

// Self-attention, B=4 C=256 H=W=64 -> N=4096, dk=C/8=32.
// Stage 1: P = [Wq;Wk;Wv] @ x  -> ws [B][320][N]  (f rows 0..31, g rows 32..63, v rows 64..319)
// Stage 2: flash attention (queries j, keys i), fp32 WMMA 16x16x4, with
//          double-buffered async global->LDS staging of the shared f/v key tiles.

#define BB     4
#define CC     256
#define NN     4096
#define DK     32
#define PROWS  320   // 32 + 32 + 256
#define NITER  (NN / 16)

typedef __attribute__((ext_vector_type(2))) float v2f;
typedef __attribute__((ext_vector_type(8))) float v8f;

static __device__ __forceinline__ v8f wmma_f32(v2f a, v2f b, v8f c) {
  return __builtin_amdgcn_wmma_f32_16x16x4_f32(false, a, false, b, (short)0, c, false, false);
}

static __device__ __forceinline__ v8f v8f_zero() {
  v8f z;
#pragma unroll
  for (int i = 0; i < 8; ++i) z[i] = 0.0f;
  return z;
}

// Issue one async 16-byte global->LDS copy (ASYNCcnt++).
static __device__ __forceinline__ void async_copy_b128(const float* src, const float* lds_dst) {
  unsigned dst = (unsigned)(size_t)lds_dst;   // LDS aperture: low 32 bits = DS offset
  asm volatile("global_load_async_to_lds_b128 %0, %1, off"
               :: "v"(dst), "v"(src) : "memory");
}

// ---------------- Stage 1: projection GEMM ----------------
// grid = (NN/64, PROWS/16, BB), block = 128 (4 waves). Each wave: one 16(m)x16(n) tile, K=256.
__global__ void __launch_bounds__(128)
proj_kernel(const float* __restrict__ x, const float* __restrict__ Wq,
            const float* __restrict__ Wk, const float* __restrict__ Wv,
            float* __restrict__ P) {
  const int lane = threadIdx.x & 31;
  const int wv   = threadIdx.x >> 5;
  const int half = lane >> 4;
  const int lm   = lane & 15;
  const int n0   = (blockIdx.x * 4 + wv) * 16;
  const int mt   = blockIdx.y;
  const int b    = blockIdx.z;

  const float* W; int mrow;
  if (mt < 2)      { W = Wq; mrow = mt * 16; }
  else if (mt < 4) { W = Wk; mrow = (mt - 2) * 16; }
  else             { W = Wv; mrow = (mt - 4) * 16; }

  const float* xb   = x + (size_t)b * CC * NN;
  const float* wrow = W + (size_t)(mrow + lm) * CC;

  v8f acc = v8f_zero();
#pragma unroll 8
  for (int kk = 0; kk < CC; kk += 4) {
    v2f a = *(const v2f*)(wrow + kk + 2 * half);                  // A[m,k] pair (b64)
    const float* xp = xb + (size_t)(kk + 2 * half) * NN + n0 + lm;
    v2f bm; bm.x = xp[0]; bm.y = xp[NN];                          // B[k,n]
    acc = wmma_f32(a, bm, acc);
  }

  float* Pb = P + ((size_t)b * PROWS + (size_t)mt * 16) * NN + n0 + lm;
#pragma unroll
  for (int r = 0; r < 8; ++r)
    Pb[(size_t)(r + 8 * half) * NN] = acc[r];
}

// ---------------- Stage 2: flash attention ----------------
// grid = (NN/64, BB), block = 128 (4 waves). Each wave: 16 queries, all 256 channels.
// f/v key tiles are shared by all 4 waves -> staged once per block via async DMA, double-buffered.
__global__ void __launch_bounds__(128)
attn_kernel(const float* __restrict__ P, float* __restrict__ out) {
  __shared__ float fT[2][DK * 16];    // f tile  [q=0..31][i=0..15], per buffer
  __shared__ float vT[2][CC * 16];    // v tile  [c=0..255][i=0..15], per buffer
  __shared__ float ldsP[4][16 * 16];  // per-wave probability tile [j][i]
  __shared__ float ldsX[4][16];       // per-wave row->column broadcast

  const int tid  = threadIdx.x;
  const int lane = tid & 31;
  const int wv   = tid >> 5;
  const int half = lane >> 4;
  const int lm   = lane & 15;
  const int b    = blockIdx.y;
  const int j0   = blockIdx.x * 64 + wv * 16;

  const float* fp = P + (size_t)b * PROWS * NN;        // [32][N]
  const float* gp = fp + (size_t)DK * NN;              // [32][N]
  const float* vp = fp + (size_t)(2 * DK) * NN;        // [256][N]
  float* myP = ldsP[wv];
  float* myX = ldsX[wv];

  // async-copy work split: thread -> (row = tid/4, col = (tid%4)*4)
  const int crow = tid >> 2;
  const int ccol = (tid & 3) * 4;

  // prologue: stage tile 0 into buffer 0 (9 x b128 per thread)
  {
    async_copy_b128(fp + (size_t)crow * NN + ccol, &fT[0][crow * 16 + ccol]);
#pragma unroll
    for (int rr = 0; rr < 8; ++rr) {
      const int c = rr * 32 + crow;
      async_copy_b128(vp + (size_t)c * NN + ccol, &vT[0][c * 16 + ccol]);
    }
  }

  // A = Q tile: A[j, q] = g[q, j0+j]; registers for whole key loop.
  v2f aG[8];
#pragma unroll
  for (int kk = 0; kk < DK; kk += 4) {
    const int q = kk + 2 * half;
    v2f a;
    a.x = gp[(size_t)q * NN + j0 + lm];
    a.y = gp[(size_t)(q + 1) * NN + j0 + lm];
    aG[kk >> 2] = a;
  }

  v8f O[16];
#pragma unroll
  for (int ct = 0; ct < 16; ++ct) O[ct] = v8f_zero();

  float m_r[8], l_r[8];
#pragma unroll
  for (int r = 0; r < 8; ++r) { m_r[r] = -__builtin_inff(); l_r[r] = 0.0f; }

  for (int it = 0; it < NITER; ++it) {
    const int ii  = it * 16;
    const int buf = it & 1;

    // prefetch next tile into the other buffer (dummy ii=0 on last iter keeps counts uniform)
    {
      const int nxt = (it + 1 < NITER) ? (ii + 16) : 0;
      async_copy_b128(fp + (size_t)crow * NN + nxt + ccol, &fT[buf ^ 1][crow * 16 + ccol]);
#pragma unroll
      for (int rr = 0; rr < 8; ++rr) {
        const int c = rr * 32 + crow;
        async_copy_b128(vp + (size_t)c * NN + nxt + ccol, &vT[buf ^ 1][c * 16 + ccol]);
      }
    }
    // 18 outstanding; <=9 left means the current tile's 9 have landed (in-order retire)
    asm volatile("s_wait_asynccnt 0x9" ::: "memory");
    __syncthreads();

    const float* fL = fT[buf];
    const float* vL = vT[buf];

    // ---- S[j, i] tile, K=32 -> 8 WMMAs; B operand from LDS f tile
    v8f s = v8f_zero();
#pragma unroll
    for (int kk = 0; kk < DK; kk += 4) {
      const int q = kk + 2 * half;
      v2f bm;
      bm.x = fL[q * 16 + lm];
      bm.y = fL[(q + 1) * 16 + lm];
      s = wmma_f32(aG[kk >> 2], bm, s);
    }

    // ---- online softmax along i (rows j = r + 8*half; 16-lane reductions)
    float sc_r[8];
#pragma unroll
    for (int r = 0; r < 8; ++r) {
      float sv   = s[r] * 0.0625f;  // 1/sqrt(256)
      float rmax = sv;
      rmax = fmaxf(rmax, __shfl_xor(rmax, 1, 32));
      rmax = fmaxf(rmax, __shfl_xor(rmax, 2, 32));
      rmax = fmaxf(rmax, __shfl_xor(rmax, 4, 32));
      rmax = fmaxf(rmax, __shfl_xor(rmax, 8, 32));
      const float mnew = fmaxf(m_r[r], rmax);
      const float p    = __expf(sv - mnew);
      float rsum = p;
      rsum += __shfl_xor(rsum, 1, 32);
      rsum += __shfl_xor(rsum, 2, 32);
      rsum += __shfl_xor(rsum, 4, 32);
      rsum += __shfl_xor(rsum, 8, 32);
      const float scl = __expf(m_r[r] - mnew);
      l_r[r] = l_r[r] * scl + rsum;
      m_r[r] = mnew;
      sc_r[r] = scl;
      s[r] = p;
    }

    if (lm == 0) {
#pragma unroll
      for (int r = 0; r < 8; ++r) myX[r + 8 * half] = sc_r[r];
    }
#pragma unroll
    for (int r = 0; r < 8; ++r) myP[(r + 8 * half) * 16 + lm] = s[r];

    const float myScale = myX[lm];
#pragma unroll
    for (int ct = 0; ct < 16; ++ct)
#pragma unroll
      for (int r = 0; r < 8; ++r) O[ct][r] *= myScale;

    // ---- O[c, j] += v[c, i] * P[j, i]; A from LDS v tile (b64), B = P^T from LDS
    v2f bP[4];
#pragma unroll
    for (int k4 = 0; k4 < 4; ++k4)
      bP[k4] = *(const v2f*)(myP + lm * 16 + k4 * 4 + 2 * half);

    const float* vrow = vL + (size_t)lm * 16 + 2 * half;
#pragma unroll
    for (int ct = 0; ct < 16; ++ct) {
      const float* va = vrow + (size_t)(ct * 16) * 16;
#pragma unroll
      for (int k4 = 0; k4 < 4; ++k4) {
        v2f aV = *(const v2f*)(va + k4 * 4);
        O[ct] = wmma_f32(aV, bP[k4], O[ct]);
      }
    }

    __syncthreads();   // all readers done before next iteration's DMA overwrites
  }

  if (lm == 0) {
#pragma unroll
    for (int r = 0; r < 8; ++r) myX[r + 8 * half] = l_r[r];
  }
  const float linv = 1.0f / myX[lm];

  float* ob = out + (size_t)b * CC * NN + j0 + lm;
#pragma unroll
  for (int ct = 0; ct < 16; ++ct)
#pragma unroll
    for (int r = 0; r < 8; ++r)
      ob[(size_t)(ct * 16 + r + 8 * half) * NN] = O[ct][r] * linv;
}

extern "C" void kernel_launch(void* const* d_in, const int* in_sizes, int n_in,
                              void* d_out, int out_size, void* d_ws, size_t ws_size,
                              hipStream_t stream) {
  (void)in_sizes; (void)n_in; (void)out_size; (void)ws_size;
  const float* x  = (const float*)d_in[0];
  const float* Wq = (const float*)d_in[1];
  const float* Wk = (const float*)d_in[2];
  const float* Wv = (const float*)d_in[3];
  float* outp = (float*)d_out;
  float* P    = (float*)d_ws;   // needs B*320*N*4 = ~21 MB

  dim3 pg(NN / 64, PROWS / 16, BB);
  proj_kernel<<<pg, 128, 0, stream>>>(x, Wq, Wk, Wv, P);

  dim3 ag(NN / 64, BB);
  attn_kernel<<<ag, 128, 0, stream>>>(P, outp);
}